// nmfl_11381663335025
// MI455X (gfx1250) — compile-verified
//
#include <hip/hip_runtime.h>

typedef __attribute__((ext_vector_type(2))) float v2f;
typedef __attribute__((ext_vector_type(8))) float v8f;

#define M_DIM 1024
#define K_DIM 32
#define N_DIM 2048
#define NNLS_ITERS 1500
#define POWER_ITERS 128

// ---------------------------------------------------------------------------
// Kernel 1: G = relu(theta)^T relu(theta)  (32x32), L = lambda_max(G) + 1e-6
// One block of 1024 threads: thread (i,j) computes G[i][j]; then a 32-lane
// power iteration (Perron-Frobenius: G > 0 entrywise, v=1 start converges)
// and a Rayleigh quotient for lambda_max. Output: ws[0..1023]=G, ws[1024]=L.
// ---------------------------------------------------------------------------
__global__ void nnls_gram_kernel(const float* __restrict__ theta,
                                 float* __restrict__ gOut) {
    __shared__ float Gs[K_DIM * K_DIM];
    __shared__ float v[K_DIM];
    __shared__ float w[K_DIM];
    __shared__ float scale_s;

    const int tid = threadIdx.x;
    const int i = tid >> 5;    // 0..31
    const int j = tid & 31;    // 0..31

    float acc = 0.0f;
    for (int m = 0; m < M_DIM; ++m) {
        float a = theta[m * K_DIM + i]; a = a > 0.0f ? a : 0.0f;
        float b = theta[m * K_DIM + j]; b = b > 0.0f ? b : 0.0f;
        acc = fmaf(a, b, acc);
    }
    Gs[i * K_DIM + j] = acc;
    gOut[i * K_DIM + j] = acc;
    if (tid < K_DIM) v[tid] = 1.0f;
    __syncthreads();

    for (int it = 0; it < POWER_ITERS; ++it) {
        if (tid < K_DIM) {
            float s = 0.0f;
            #pragma unroll 4
            for (int c = 0; c < K_DIM; ++c) s = fmaf(Gs[tid * K_DIM + c], v[c], s);
            w[tid] = s;
        }
        __syncthreads();
        if (tid == 0) {
            float nrm = 0.0f;
            for (int c = 0; c < K_DIM; ++c) nrm = fmaf(w[c], w[c], nrm);
            scale_s = rsqrtf(nrm);
        }
        __syncthreads();
        if (tid < K_DIM) v[tid] = w[tid] * scale_s;
        __syncthreads();
    }

    if (tid == 0) {
        // Rayleigh quotient with unit v: lambda = v^T G v
        float lam = 0.0f;
        for (int r = 0; r < K_DIM; ++r) {
            float s = 0.0f;
            for (int c = 0; c < K_DIM; ++c) s = fmaf(Gs[r * K_DIM + c], v[c], s);
            lam = fmaf(v[r], s, lam);
        }
        gOut[K_DIM * K_DIM] = lam + 1e-6f;
    }
}

// ---------------------------------------------------------------------------
// Kernel 2: per-block NNLS on a 32x16 column tile of B. 1 wave per block.
// B update: B <- relu( (B + T/L) + (-G/L) @ B ), computed with f32 WMMA
// 16x16x4 (two 16-row output blocks x 8 K-chunks = 16 WMMA per iteration).
// A-operand (-G/L) is loop-invariant in registers; B round-trips through a
// padded LDS tile to convert C/D layout -> B-operand layout each iteration.
// ---------------------------------------------------------------------------
__global__ void __launch_bounds__(32)
nnls_pgd_kernel(const float* __restrict__ X,
                const float* __restrict__ theta,
                const float* __restrict__ gIn,
                float* __restrict__ out) {
    __shared__ float Bt[K_DIM][17];  // [row][col], padded to dodge bank conflicts

    const int lane = threadIdx.x;    // 0..31 (wave32)
    const int col  = lane & 15;      // output column within tile / A-row
    const int hi   = lane >> 4;      // lane half
    const int nbase = blockIdx.x * 16;

    const float L = gIn[K_DIM * K_DIM];
    const float invL = 1.0f / L;
    const float negInvL = -invL;

    // --- A-operand: four 16x16 blocks of (-G/L), each as 4 K-chunks of 16x4.
    // f32 A 16x4 layout: lanes 0-15 hold M=lane, K=0,1; lanes 16-31 M=lane-16, K=2,3.
    v2f A0[8], A1[8];
    #pragma unroll
    for (int c = 0; c < 8; ++c) {
        const int ks = 4 * c + 2 * hi;
        A0[c].x = gIn[col * K_DIM + ks]     * negInvL;
        A0[c].y = gIn[col * K_DIM + ks + 1] * negInvL;
        A1[c].x = gIn[(col + 16) * K_DIM + ks]     * negInvL;
        A1[c].y = gIn[(col + 16) * K_DIM + ks + 1] * negInvL;
    }

    // --- T tile scaled by 1/L, directly in C/D layout:
    // TL0[j] = T[j+8*hi][nbase+col]/L ; TL1[j] = T[j+8*hi+16][nbase+col]/L
    v8f TL0 = {}, TL1 = {};
    for (int m = 0; m < M_DIM; ++m) {
        const float x = X[m * N_DIM + nbase + col];
        const float* trow = theta + m * K_DIM + 8 * hi;
        #pragma unroll
        for (int j = 0; j < 8; ++j) {
            float t0 = trow[j];      t0 = t0 > 0.0f ? t0 : 0.0f;
            float t1 = trow[j + 16]; t1 = t1 > 0.0f ? t1 : 0.0f;
            TL0[j] = fmaf(t0, x, TL0[j]);
            TL1[j] = fmaf(t1, x, TL1[j]);
        }
    }
    #pragma unroll
    for (int j = 0; j < 8; ++j) { TL0[j] *= invL; TL1[j] *= invL; }

    // --- B state: C/D layout registers + mirror in LDS (for B-operand loads).
    v8f B0 = {}, B1 = {};
    #pragma unroll
    for (int j = 0; j < 8; ++j) {
        Bt[j + 8 * hi][col] = 0.0f;
        Bt[j + 8 * hi + 16][col] = 0.0f;
    }
    __syncthreads();  // single-wave WG: compiler fence + near-NOP barrier

    for (int it = 0; it < NNLS_ITERS; ++it) {
        // B-operand chunks (4x16 f32 layout: V0 = K(4c)/K(4c+2), V1 = K(4c+1)/K(4c+3))
        v2f bch[8];
        #pragma unroll
        for (int c = 0; c < 8; ++c) {
            const int k = 4 * c + 2 * hi;
            bch[c].x = Bt[k][col];
            bch[c].y = Bt[k + 1][col];
        }

        // C operand: B + T/L
        v8f acc0, acc1;
        #pragma unroll
        for (int j = 0; j < 8; ++j) {
            acc0[j] = B0[j] + TL0[j];
            acc1[j] = B1[j] + TL1[j];
        }

        // acc += (-G/L) @ B   (K=32 via 8 chunks of K=4, two independent chains)
        #pragma unroll
        for (int c = 0; c < 8; ++c) {
            acc0 = __builtin_amdgcn_wmma_f32_16x16x4_f32(
                false, A0[c], false, bch[c], (short)0, acc0, false, false);
        }
        #pragma unroll
        for (int c = 0; c < 8; ++c) {
            acc1 = __builtin_amdgcn_wmma_f32_16x16x4_f32(
                false, A1[c], false, bch[c], (short)0, acc1, false, false);
        }

        // Projection onto nonnegative orthant
        #pragma unroll
        for (int j = 0; j < 8; ++j) {
            B0[j] = acc0[j] > 0.0f ? acc0[j] : 0.0f;
            B1[j] = acc1[j] > 0.0f ? acc1[j] : 0.0f;
        }

        __syncthreads();  // old-B reads done before overwrite
        #pragma unroll
        for (int j = 0; j < 8; ++j) {
            Bt[j + 8 * hi][col]      = B0[j];
            Bt[j + 8 * hi + 16][col] = B1[j];
        }
        __syncthreads();  // new B visible for next iteration's loads
    }

    // beta is (32, 2048) row-major
    #pragma unroll
    for (int j = 0; j < 8; ++j) {
        out[(j + 8 * hi)      * N_DIM + nbase + col] = B0[j];
        out[(j + 8 * hi + 16) * N_DIM + nbase + col] = B1[j];
    }
}

extern "C" void kernel_launch(void* const* d_in, const int* in_sizes, int n_in,
                              void* d_out, int out_size, void* d_ws, size_t ws_size,
                              hipStream_t stream) {
    const float* X     = (const float*)d_in[0];  // (1024, 2048)
    const float* theta = (const float*)d_in[1];  // (1024, 32)
    float* gws = (float*)d_ws;                   // 1025 floats: G (32x32) + L

    nnls_gram_kernel<<<1, 1024, 0, stream>>>(theta, gws);
    nnls_pgd_kernel<<<N_DIM / 16, 32, 0, stream>>>(X, theta, gws, (float*)d_out);
}